// Block_41515153883628
// MI455X (gfx1250) — compile-verified
//
#include <hip/hip_runtime.h>
#include <math.h>

// ---------------------------------------------------------------------------
// Hyena block for MI455X (gfx1250, wave32).
// GEMMs: V_WMMA_F32_16X16X4_F32 (full fp32 on matrix cores), 128x64 block
//        tile, 16x64 wave tile (4 accumulators), double-buffered LDS filled
//        with GLOBAL_LOAD_ASYNC_TO_LDS_B128 (ASYNCcnt).
// FFT long-conv: per-channel radix-2 length-4096 FFT in LDS.
// ---------------------------------------------------------------------------

typedef __attribute__((ext_vector_type(2))) float v2f;
typedef __attribute__((ext_vector_type(8))) float v8f;

#define LN_EPS 1e-5f

// dims (fixed by the reference problem)
#define BB 4
#define LL 2048
#define DD 768
#define HH 3072
#define NTOK (BB * LL)        // 8192
#define FFTN 4096
#define FFTLOG 12

// LDS tile row stride: 36 floats = 144 B (16B-aligned rows for async B128
// stores, and 36*m mod 64 distinct for m=0..15 -> conflict-free frag reads)
#define TSTRIDE 36

// ---------------------------------------------------------------------------
// async global->LDS copy helpers (CDNA5, tracked by ASYNCcnt)
// generic pointer to __shared__ truncates to the wave-relative LDS offset
// ---------------------------------------------------------------------------
__device__ __forceinline__ unsigned lds_off(const void* p) {
    return (unsigned)(size_t)p;
}

__device__ __forceinline__ void async_load_b128(const float* gptr, void* lptr) {
    unsigned loff = lds_off(lptr);
    asm volatile("global_load_async_to_lds_b128 %0, %1, off"
                 :: "v"(loff), "v"(gptr) : "memory");
}

__device__ __forceinline__ void async_load_b32(const float* gptr, void* lptr) {
    unsigned loff = lds_off(lptr);
    asm volatile("global_load_async_to_lds_b32 %0, %1, off"
                 :: "v"(loff), "v"(gptr) : "memory");
}

__device__ __forceinline__ void wait_async() {
    asm volatile("s_wait_asynccnt 0x0" ::: "memory");
}

// ---------------------------------------------------------------------------
// block-wide sum reduction (256 threads)
// ---------------------------------------------------------------------------
__device__ __forceinline__ float block_reduce_sum(float v, float* red) {
    int tid = threadIdx.x;
    red[tid] = v;
    __syncthreads();
    #pragma unroll
    for (int off = 128; off > 0; off >>= 1) {
        if (tid < off) red[tid] += red[tid + off];
        __syncthreads();
    }
    float r = red[0];
    __syncthreads();
    return r;
}

// ---------------------------------------------------------------------------
// LayerNorm: one block per row, D = 768, 256 threads (3 elems/thread)
// ---------------------------------------------------------------------------
__global__ __launch_bounds__(256) void ln_kernel(
    const float* __restrict__ x, const float* __restrict__ g,
    const float* __restrict__ b, float* __restrict__ out) {
    __shared__ float red[256];
    const int row = blockIdx.x;
    const int tid = threadIdx.x;
    const float* xr = x + (size_t)row * DD;
    float v0 = xr[tid], v1 = xr[tid + 256], v2 = xr[tid + 512];
    float s = block_reduce_sum(v0 + v1 + v2, red);
    float mu = s * (1.0f / DD);
    float d0 = v0 - mu, d1 = v1 - mu, d2 = v2 - mu;
    float q = block_reduce_sum(d0 * d0 + d1 * d1 + d2 * d2, red);
    float rstd = rsqrtf(q * (1.0f / DD) + LN_EPS);
    float* orow = out + (size_t)row * DD;
    orow[tid]       = d0 * rstd * g[tid]       + b[tid];
    orow[tid + 256] = d1 * rstd * g[tid + 256] + b[tid + 256];
    orow[tid + 512] = d2 * rstd * g[tid + 512] + b[tid + 512];
}

// fused LN(LN(x, g2, b2), g3, b3)
__global__ __launch_bounds__(256) void ln_dual_kernel(
    const float* __restrict__ x,
    const float* __restrict__ g2, const float* __restrict__ b2,
    const float* __restrict__ g3, const float* __restrict__ b3,
    float* __restrict__ out) {
    __shared__ float red[256];
    const int row = blockIdx.x;
    const int tid = threadIdx.x;
    const float* xr = x + (size_t)row * DD;
    float v0 = xr[tid], v1 = xr[tid + 256], v2 = xr[tid + 512];
    float s = block_reduce_sum(v0 + v1 + v2, red);
    float mu = s * (1.0f / DD);
    float d0 = v0 - mu, d1 = v1 - mu, d2 = v2 - mu;
    float q = block_reduce_sum(d0 * d0 + d1 * d1 + d2 * d2, red);
    float rstd = rsqrtf(q * (1.0f / DD) + LN_EPS);
    float t0 = d0 * rstd * g2[tid]       + b2[tid];
    float t1 = d1 * rstd * g2[tid + 256] + b2[tid + 256];
    float t2 = d2 * rstd * g2[tid + 512] + b2[tid + 512];
    // second LN
    float s2 = block_reduce_sum(t0 + t1 + t2, red);
    float mu2 = s2 * (1.0f / DD);
    float e0 = t0 - mu2, e1 = t1 - mu2, e2 = t2 - mu2;
    float q2 = block_reduce_sum(e0 * e0 + e1 * e1 + e2 * e2, red);
    float rstd2 = rsqrtf(q2 * (1.0f / DD) + LN_EPS);
    float* orow = out + (size_t)row * DD;
    orow[tid]       = e0 * rstd2 * g3[tid]       + b3[tid];
    orow[tid + 256] = e1 * rstd2 * g3[tid + 256] + b3[tid + 256];
    orow[tid + 512] = e2 * rstd2 * g3[tid + 512] + b3[tid + 512];
}

// ---------------------------------------------------------------------------
// GEMM: out[m,n] = sum_k A[m,k] * W[n,k] + bias[n]   (+ epilogue)
// A: [M,K] row-major, W: [N,K] row-major.
// Block tile 128x64, 8 waves; wave w owns rows w*16..w*16+15, all 64 cols
// (four 16x16 accumulators -> each A fragment feeds 4 WMMAs).
// K staged in 32-wide, double-buffered LDS blocks filled by async
// global->LDS B128 copies; inner step = fp32 WMMA 16x16x4.
// ---------------------------------------------------------------------------
#define EPI_BIAS 0
#define EPI_GELU 1
#define EPI_RES  2

template<int EPI>
__global__ __launch_bounds__(256) void gemm_wmma_kernel(
    const float* __restrict__ A, const float* __restrict__ W,
    const float* __restrict__ bias, const float* __restrict__ res,
    float* __restrict__ out, int M, int N, int K) {
    __shared__ float Alds[2][128][TSTRIDE];
    __shared__ float Blds[2][64][TSTRIDE];

    const int bm = blockIdx.y * 128;
    const int bn = blockIdx.x * 64;
    const int tid = threadIdx.x;
    const int wave = tid >> 5;       // 0..7 -> row stripe
    const int lane = tid & 31;

    // per-thread cooperative load slots (B128 chunks, 256 threads):
    // A tile 128x32 = 1024 chunks (4/thread), B tile 64x32 = 512 chunks (2/thread)
    const int l_row0 = tid >> 3;                 // 0..31
    const int l_col  = (tid & 7) << 2;           // 0..28 step 4

    v8f c[4] = {v8f{}, v8f{}, v8f{}, v8f{}};

    // prefetch first K-block into buffer 0
    {
        #pragma unroll
        for (int r = 0; r < 4; ++r) {
            async_load_b128(A + (size_t)(bm + l_row0 + 32 * r) * K + l_col,
                            &Alds[0][l_row0 + 32 * r][l_col]);
        }
        #pragma unroll
        for (int r = 0; r < 2; ++r) {
            async_load_b128(W + (size_t)(bn + l_row0 + 32 * r) * K + l_col,
                            &Blds[0][l_row0 + 32 * r][l_col]);
        }
    }
    wait_async();
    __syncthreads();

    const int mrow = wave * 16 + (lane & 15);
    const int nrow = lane & 15;
    const int kh   = (lane >> 4) << 1;   // 0 for lanes 0-15, 2 for 16-31

    int buf = 0;
    for (int k0 = 0; k0 < K; k0 += 32) {
        // prefetch next K-block into the other buffer (overlaps with WMMAs)
        if (k0 + 32 < K) {
            const int kn = k0 + 32;
            #pragma unroll
            for (int r = 0; r < 4; ++r) {
                async_load_b128(A + (size_t)(bm + l_row0 + 32 * r) * K + kn + l_col,
                                &Alds[buf ^ 1][l_row0 + 32 * r][l_col]);
            }
            #pragma unroll
            for (int r = 0; r < 2; ++r) {
                async_load_b128(W + (size_t)(bn + l_row0 + 32 * r) * K + kn + l_col,
                                &Blds[buf ^ 1][l_row0 + 32 * r][l_col]);
            }
        }

        #pragma unroll
        for (int kk = 0; kk < 32; kk += 4) {
            int koff = kk + kh;
            v2f a;
            a.x = Alds[buf][mrow][koff];
            a.y = Alds[buf][mrow][koff + 1];
            #pragma unroll
            for (int nt = 0; nt < 4; ++nt) {
                v2f b;
                b.x = Blds[buf][nrow + 16 * nt][koff];
                b.y = Blds[buf][nrow + 16 * nt][koff + 1];
                c[nt] = __builtin_amdgcn_wmma_f32_16x16x4_f32(
                    false, a, false, b, (short)0, c[nt], false, false);
            }
        }

        wait_async();        // my prefetch copies have landed in LDS
        __syncthreads();     // everyone's landed; everyone done reading buf
        buf ^= 1;
    }

    // epilogue: VGPR i -> M = i + (lane<16 ? 0 : 8), N = lane & 15
    const int lrow = (lane >> 4) << 3;
    const int lcol = lane & 15;
    #pragma unroll
    for (int i = 0; i < 8; ++i) {
        int gr = bm + wave * 16 + lrow + i;
        #pragma unroll
        for (int nt = 0; nt < 4; ++nt) {
            int gc = bn + nt * 16 + lcol;
            float v = c[nt][i] + bias[gc];
            if (EPI == EPI_GELU) {
                v = 0.5f * v * (1.0f + erff(v * 0.70710678118654752f));
            } else if (EPI == EPI_RES) {
                v += res[(size_t)gr * N + gc];
            }
            out[(size_t)gr * N + gc] = v;
        }
    }
}

// ---------------------------------------------------------------------------
// radix-2 in-place DIT FFT over LDS (N = 4096), 256 threads cooperate
// ---------------------------------------------------------------------------
__device__ void fft4096(float* re, float* im, int tid, int nthr) {
    // bit-reversal permutation
    for (int i = tid; i < FFTN; i += nthr) {
        int j = (int)(__brev((unsigned)i) >> (32 - FFTLOG));
        if (j > i) {
            float tr = re[i]; re[i] = re[j]; re[j] = tr;
            float ti = im[i]; im[i] = im[j]; im[j] = ti;
        }
    }
    __syncthreads();
    for (int s = 1; s <= FFTLOG; ++s) {
        const int len = 1 << s;
        const int half = len >> 1;
        const float ang = -6.283185307179586f / (float)len;
        for (int k = tid; k < (FFTN >> 1); k += nthr) {
            int j = k & (half - 1);
            int i1 = ((k >> (s - 1)) << s) + j;
            int i2 = i1 + half;
            float wi, wr;
            __sincosf(ang * (float)j, &wi, &wr);
            float xr = re[i2], xi = im[i2];
            float tr = xr * wr - xi * wi;
            float ti = xr * wi + xi * wr;
            float ur = re[i1], ui = im[i1];
            re[i1] = ur + tr; im[i1] = ui + ti;
            re[i2] = ur - tr; im[i2] = ui - ti;
        }
        __syncthreads();
    }
}

// ---------------------------------------------------------------------------
// gated FFT conv for one order:
//   vin  = vprev * gate                               (elementwise)
//   vout = irfft(rfft(pad(vin)) * rfft(pad(filt)))[:L] + vin * fbias
// one workgroup per (b, d) channel; everything in LDS (64 KB dyn shared).
// filter taps pulled in via async global->LDS B32 copies.
// ---------------------------------------------------------------------------
__global__ __launch_bounds__(256) void fftconv_kernel(
    const float* __restrict__ vprev, int ld_prev,
    const float* __restrict__ gate, int ld_gate,
    const float* __restrict__ filt,   // [L, D] for this order
    const float* __restrict__ fbias,  // [D]   for this order
    float* __restrict__ vout) {       // [B, L, D]
    extern __shared__ float smem[];
    float* re  = smem;
    float* im  = smem + FFTN;
    float* fre = smem + 2 * FFTN;
    float* fim = smem + 3 * FFTN;

    const int d   = blockIdx.x;
    const int b   = blockIdx.y;
    const int tid = threadIdx.x;
    const int nthr = blockDim.x;

    const float* vp = vprev + (size_t)b * LL * ld_prev + d;
    const float* gp = gate  + (size_t)b * LL * ld_gate + d;

    // filter taps straight to LDS (async, no VGPR round-trip)
    for (int l = tid; l < LL; l += nthr) {
        async_load_b32(filt + (size_t)l * DD + d, &fre[l]);
    }
    // gated signal needs a VALU multiply -> normal path
    for (int l = tid; l < FFTN; l += nthr) {
        if (l < LL) {
            re[l] = vp[(size_t)l * ld_prev] * gp[(size_t)l * ld_gate];
        } else {
            re[l] = 0.0f; fre[l] = 0.0f;
        }
        im[l] = 0.0f; fim[l] = 0.0f;
    }
    wait_async();
    __syncthreads();

    fft4096(re, im, tid, nthr);
    fft4096(fre, fim, tid, nthr);

    // pointwise product + conjugate (prepares inverse-by-forward-FFT)
    for (int i = tid; i < FFTN; i += nthr) {
        float ar = re[i], ai = im[i];
        float br = fre[i], bi = fim[i];
        float pr = ar * br - ai * bi;
        float pi = ar * bi + ai * br;
        re[i] = pr; im[i] = -pi;
    }
    __syncthreads();

    fft4096(re, im, tid, nthr);

    const float inv = 1.0f / (float)FFTN;
    const float fb = fbias[d];
    float* vo = vout + (size_t)b * LL * DD + d;
    for (int l = tid; l < LL; l += nthr) {
        float vin = vp[(size_t)l * ld_prev] * gp[(size_t)l * ld_gate];
        vo[(size_t)l * DD] = re[l] * inv + vin * fb;
    }
}

// ---------------------------------------------------------------------------
// launcher
// ---------------------------------------------------------------------------
extern "C" void kernel_launch(void* const* d_in, const int* in_sizes, int n_in,
                              void* d_out, int out_size, void* d_ws, size_t ws_size,
                              hipStream_t stream) {
    (void)in_sizes; (void)n_in; (void)out_size; (void)ws_size;

    const float* x      = (const float*)d_in[0];
    const float* ln1_g  = (const float*)d_in[1];
    const float* ln1_b  = (const float*)d_in[2];
    const float* in_W   = (const float*)d_in[3];   // [3*D, D]
    const float* in_b   = (const float*)d_in[4];
    const float* filters= (const float*)d_in[5];   // [2, L, D]
    const float* fbias  = (const float*)d_in[6];   // [2, D]
    const float* out_W  = (const float*)d_in[7];   // [D, D]
    const float* out_b  = (const float*)d_in[8];
    const float* ln2_g  = (const float*)d_in[9];
    const float* ln2_b  = (const float*)d_in[10];
    const float* ln3_g  = (const float*)d_in[11];
    const float* ln3_b  = (const float*)d_in[12];
    const float* fc1_W  = (const float*)d_in[13];  // [H, D]
    const float* fc1_b  = (const float*)d_in[14];
    const float* fc2_W  = (const float*)d_in[15];  // [D, H]
    const float* fc2_b  = (const float*)d_in[16];
    float* y = (float*)d_out;

    // workspace layout (floats)
    const size_t NM = (size_t)NTOK * DD;           // 6.29M floats
    float* ws   = (float*)d_ws;
    float* h    = ws;                  // LN1 out; later reused: v1, h2
    float* z    = h + NM;              // [N, 3*D]
    float* v2   = z + 3 * NM;          // [N, D]
    float* xmid = v2 + NM;             // [N, D]
    float* gbuf = xmid + NM;           // [N, H]
    float* v1 = h;                     // reuse (h dead after in_proj)
    float* h2 = h;                     // reuse (v1 dead after out_proj)

    const dim3 blk(256);
    const size_t conv_lds = (size_t)4 * FFTN * sizeof(float); // 64 KB

    // 1) h = LN1(x)
    ln_kernel<<<NTOK, blk, 0, stream>>>(x, ln1_g, ln1_b, h);

    // 2) z = h @ in_W^T + in_b           [8192 x 2304]
    gemm_wmma_kernel<EPI_BIAS><<<dim3((3 * DD) / 64, NTOK / 128), blk, 0, stream>>>(
        h, in_W, in_b, nullptr, z, NTOK, 3 * DD, DD);

    // 3) order 0: v1 = fftconv(parts[2] * parts[0], filters[0]) ; gate offset 0
    fftconv_kernel<<<dim3(DD, BB), blk, conv_lds, stream>>>(
        z + 2 * DD, 3 * DD, z, 3 * DD,
        filters, fbias, v1);

    // 4) order 1: v2 = fftconv(v1 * parts[1], filters[1])
    fftconv_kernel<<<dim3(DD, BB), blk, conv_lds, stream>>>(
        v1, DD, z + DD, 3 * DD,
        filters + (size_t)LL * DD, fbias + DD, v2);

    // 5) xmid = v2 @ out_W^T + out_b + x  [8192 x 768]
    gemm_wmma_kernel<EPI_RES><<<dim3(DD / 64, NTOK / 128), blk, 0, stream>>>(
        v2, out_W, out_b, x, xmid, NTOK, DD, DD);

    // 6) h2 = LN3(LN2(xmid))
    ln_dual_kernel<<<NTOK, blk, 0, stream>>>(xmid, ln2_g, ln2_b, ln3_g, ln3_b, h2);

    // 7) gbuf = gelu(h2 @ fc1_W^T + fc1_b)  [8192 x 3072]
    gemm_wmma_kernel<EPI_GELU><<<dim3(HH / 64, NTOK / 128), blk, 0, stream>>>(
        h2, fc1_W, fc1_b, nullptr, gbuf, NTOK, HH, DD);

    // 8) y = gbuf @ fc2_W^T + fc2_b + xmid  [8192 x 768]
    gemm_wmma_kernel<EPI_RES><<<dim3(DD / 64, NTOK / 128), blk, 0, stream>>>(
        gbuf, fc2_W, fc2_b, xmid, y, NTOK, DD, HH);
}